// Encoder_18717467476687
// MI455X (gfx1250) — compile-verified
//
#include <hip/hip_runtime.h>

typedef __attribute__((ext_vector_type(2))) float v2f;
typedef __attribute__((ext_vector_type(8))) float v8f;

#define IN_CH 128
#define HID   128

// ---------------- K0: deg = 1 (self loop) ----------------
__global__ void k_init_deg(unsigned* __restrict__ deg, int N) {
  int i = blockIdx.x * blockDim.x + threadIdx.x;
  if (i < N) deg[i] = 1u;
}

// ---------------- K1: in-degree count on dst ----------------
__global__ void k_count(const int* __restrict__ ei, int E, unsigned* __restrict__ deg) {
  int e = blockIdx.x * blockDim.x + threadIdx.x;
  if (e < E) atomicAdd(&deg[ei[E + e]], 1u);
}

// ---------------- K2: dis = rsqrt(deg) ----------------
__global__ void k_dis(const unsigned* __restrict__ deg, float* __restrict__ dis, int N) {
  int i = blockIdx.x * blockDim.x + threadIdx.x;
  if (i < N) dis[i] = rsqrtf((float)deg[i]);
}

// ---------------- K3: h = x @ W via V_WMMA_F32_16X16X4_F32 ----------------
// One wave computes one 16x16 output tile. blockDim = 128 (4 waves),
// grid = (2, N/16): col tile = blockIdx.x*4 + wave (8 col tiles of 16).
__global__ __launch_bounds__(128) void k_gemm(const float* __restrict__ x,
                                              const float* __restrict__ W,
                                              float* __restrict__ h) {
  const int lane = threadIdx.x & 31;
  const int wave = threadIdx.x >> 5;
  const int row0 = blockIdx.y * 16;
  const int col0 = (blockIdx.x * 4 + wave) * 16;

  const int mn  = lane & 15;            // M for A, N for B/C
  const int kof = (lane >> 4) << 1;     // lanes 0-15: K+0/K+1, lanes 16-31: K+2/K+3

  v8f acc = {};
  const float* xr = x + (size_t)(row0 + mn) * IN_CH;
  const float* Wc = W + col0 + mn;

#pragma unroll
  for (int k = 0; k < IN_CH; k += 4) {
    // A: 16x4 f32.  VGPR0 = K=(k+kof), VGPR1 = K=(k+kof+1), contiguous -> float2 load.
    const float2 av = *(const float2*)(xr + k + kof);
    v2f a; a[0] = av.x; a[1] = av.y;
    // B: 4x16 f32.  VGPR0 = row K=(k+kof), VGPR1 = row K=(k+kof+1), N = lane&15.
    v2f b;
    b[0] = Wc[(size_t)(k + kof) * HID];
    b[1] = Wc[(size_t)(k + kof + 1) * HID];
    acc = __builtin_amdgcn_wmma_f32_16x16x4_f32(false, a, false, b,
                                                (short)0, acc, false, false);
  }

  // C/D layout: VGPR r -> M = r (lanes 0-15) / M = r+8 (lanes 16-31), N = lane&15.
  const int mo = (lane >> 4) * 8;
  float* hr = h + (size_t)(row0 + mo) * HID + col0 + mn;
#pragma unroll
  for (int r = 0; r < 8; ++r)
    hr[(size_t)r * HID] = acc[r];
}

// ---------------- K4: out = b + dis^2 * h  (self-loop + bias) ----------------
__global__ void k_init_out(const float* __restrict__ h, const float* __restrict__ dis,
                           const float* __restrict__ bias, float* __restrict__ out, int N) {
  int t = blockIdx.x * blockDim.x + threadIdx.x;   // over N*32 float4 groups
  if (t >= N * 32) return;
  int i  = t >> 5;
  int cg = (t & 31) << 2;
  float nrm = dis[i] * dis[i];
  float4 hv = *(const float4*)(h + (size_t)i * HID + cg);
  float4 bv = *(const float4*)(bias + cg);
  float4 o;
  o.x = bv.x + nrm * hv.x;
  o.y = bv.y + nrm * hv.y;
  o.z = bv.z + nrm * hv.z;
  o.w = bv.w + nrm * hv.w;
  *(float4*)(out + (size_t)i * HID + cg) = o;
}

// ---------------- K5: edge scatter, wave-per-edge, f32 HW atomics ----------------
__global__ __launch_bounds__(256) void k_scatter(const int* __restrict__ ei, int E,
                                                 const float* __restrict__ h,
                                                 const float* __restrict__ dis,
                                                 float* __restrict__ out) {
  int w    = (blockIdx.x * blockDim.x + threadIdx.x) >> 5;
  int lane = threadIdx.x & 31;
  if (w >= E) return;
  int s = ei[w];
  int d = ei[E + w];
  float nrm = dis[s] * dis[d];
  const float4 hv = *(const float4*)(h + (size_t)s * HID + (lane << 2));
  float* op = out + (size_t)d * HID + (lane << 2);
  unsafeAtomicAdd(op + 0, nrm * hv.x);
  unsafeAtomicAdd(op + 1, nrm * hv.y);
  unsafeAtomicAdd(op + 2, nrm * hv.z);
  unsafeAtomicAdd(op + 3, nrm * hv.w);
}

// ---------------- K6: in-place per-channel PReLU ----------------
__global__ void k_prelu(float* __restrict__ out, const float* __restrict__ pw, int N) {
  int t = blockIdx.x * blockDim.x + threadIdx.x;   // over N*32 float4 groups
  if (t >= N * 32) return;
  int cg = (t & 31) << 2;
  float4 v = *(float4*)(out + (size_t)t * 4);
  float4 p = *(const float4*)(pw + cg);
  v.x = v.x >= 0.f ? v.x : p.x * v.x;
  v.y = v.y >= 0.f ? v.y : p.y * v.y;
  v.z = v.z >= 0.f ? v.z : p.z * v.z;
  v.w = v.w >= 0.f ? v.w : p.w * v.w;
  *(float4*)(out + (size_t)t * 4) = v;
}

extern "C" void kernel_launch(void* const* d_in, const int* in_sizes, int n_in,
                              void* d_out, int out_size, void* d_ws, size_t ws_size,
                              hipStream_t stream) {
  const float* x  = (const float*)d_in[0];
  const int*   ei = (const int*)d_in[1];
  const float* W  = (const float*)d_in[2];
  const float* b  = (const float*)d_in[3];
  const float* pw = (const float*)d_in[4];
  float* out = (float*)d_out;

  const int N = in_sizes[0] / IN_CH;   // 100000
  const int E = in_sizes[1] / 2;       // 600000

  // workspace layout: deg[N u32] | dis[N f32] | h[N*HID f32]
  char* ws = (char*)d_ws;
  size_t o0 = 0;
  size_t o1 = (o0 + (size_t)N * 4 + 255) & ~(size_t)255;
  size_t o2 = (o1 + (size_t)N * 4 + 255) & ~(size_t)255;
  unsigned* deg = (unsigned*)(ws + o0);
  float*    dis = (float*)(ws + o1);
  float*    h   = (float*)(ws + o2);

  const int T = 256;
  k_init_deg<<<(N + T - 1) / T, T, 0, stream>>>(deg, N);
  k_count  <<<(E + T - 1) / T, T, 0, stream>>>(ei, E, deg);
  k_dis    <<<(N + T - 1) / T, T, 0, stream>>>(deg, dis, N);

  dim3 ggrid(2, N / 16);               // 8 col tiles x (N/16) row tiles, 4 waves/block
  k_gemm<<<ggrid, 128, 0, stream>>>(x, W, h);

  int tq = N * 32;                     // float4 groups
  k_init_out<<<(tq + T - 1) / T, T, 0, stream>>>(h, dis, b, out, N);

  int sblocks = (E + 7) / 8;           // 8 waves (edges) per 256-thread block
  k_scatter<<<sblocks, T, 0, stream>>>(ei, E, h, dis, out);

  k_prelu<<<(tq + T - 1) / T, T, 0, stream>>>(out, pw, N);
}